// LCAMatrixModel_81355270521443
// MI455X (gfx1250) — compile-verified
//
#include <hip/hip_runtime.h>
#include <hip/hip_bf16.h>
#include <math.h>

// ---------------------------------------------------------------------------
// LCAMatrixModel for MI455X (gfx1250, wave32)
//   z  = x @ W_enc^T + b_enc                      (1536 x 128)
//   t  = selu(z @ W1a^T + b1a)                    (1536 x 64)
//   hh = selu(t @ W1b^T + b1b)                    (1536 x 128)
//   g  = hh @ W2a^T                               (1536 x 64)
//   o[i,j] = hardsigmoid( sum_h w2b[h]*selu(g[i,h]+g[j,h]+b2a[h]) + b2b )
// GEMMs run on V_WMMA_F32_16X16X4_F32 (fp32 matrix pipe, exact math).
// Pairwise part is VALU/exp bound; exploit o[i,j]==o[j,i] to halve work.
// ---------------------------------------------------------------------------

typedef __attribute__((ext_vector_type(2))) float v2f;
typedef __attribute__((ext_vector_type(8))) float v8f;

#define SELU_SCALE 1.0507009873554805f
#define SELU_ALPHA 1.6732632423543772f

// One wave (32 threads) computes a 16x16 tile of Out = act(A @ W^T + bias).
// A: M x K row-major, W: Nout x K row-major (so B = W^T is K x Nout).
// ACT: 0 = identity, 1 = SELU.
//
// fp32 WMMA lane layout (CDNA5 ISA):
//   A 16x4 : lane l -> M = l&15 ; VGPR0/1 hold K = (l>=16 ? 2:0) + {0,1}
//   B 4x16 : lane l -> N = l&15 ; VGPR0/1 hold K = (l>=16 ? 2:0) + {0,1}
//   C/D    : VGPR r, lanes 0-15 -> (M=r, N=lane); lanes 16-31 -> (M=r+8, N=lane-16)
template <int ACT>
__global__ void wmma_gemm_nt(const float* __restrict__ A,
                             const float* __restrict__ W,
                             const float* __restrict__ bias,
                             float* __restrict__ Out,
                             int M, int K, int Nout) {
  const int lane = threadIdx.x;            // 0..31, one wave per block
  const int row0 = blockIdx.x * 16;
  const int col0 = blockIdx.y * 16;
  const int m    = lane & 15;
  const int koff = (lane >> 4) << 1;       // 0 for lanes 0-15, 2 for lanes 16-31

  const float* __restrict__ ap = A + (size_t)(row0 + m) * K + koff;
  const float* __restrict__ bp = W + (size_t)(col0 + m) * K + koff;

  v8f acc = {};
#pragma unroll 4
  for (int k = 0; k < K; k += 4) {
    v2f a = *(const v2f*)(ap + k);         // A[row0+m][k+koff .. +1]
    v2f b = *(const v2f*)(bp + k);         // W[col0+m][k+koff .. +1] == B^T row
    acc = __builtin_amdgcn_wmma_f32_16x16x4_f32(
        /*neg_a=*/false, a, /*neg_b=*/false, b,
        /*c_mod=*/(short)0, acc, /*reuse_a=*/false, /*reuse_b=*/false);
  }

  // Epilogue: bias is per output column -> depends only on lane.
  const int   col  = col0 + m;
  const float bv   = bias ? bias[col] : 0.0f;
  const int   rbase = row0 + ((lane >> 4) << 3);   // +8 rows for upper half-wave
#pragma unroll
  for (int r = 0; r < 8; ++r) {
    float v = acc[r] + bv;
    if (ACT == 1) {
      v = SELU_SCALE * (v > 0.0f ? v : SELU_ALPHA * (__expf(v) - 1.0f));
    }
    Out[(size_t)(rbase + r) * Nout + col] = v;
  }
}

// 256 threads compute a 16x16 (i,j) output tile; H = 64.
// Symmetry: only blocks with blockIdx.y >= blockIdx.x compute; they write the
// mirrored tile as well (identical values on the diagonal -> deterministic).
__global__ void pairwise_kernel(const float* __restrict__ g,
                                const float* __restrict__ b2a,
                                const float* __restrict__ w2b,
                                const float* __restrict__ b2b,
                                float* __restrict__ out, int n) {
  if (blockIdx.y < blockIdx.x) return;     // lower triangle mirrored from upper

  __shared__ float gi[16][65];             // pad to 65: avoid 64-bank conflicts
  __shared__ float gj[16][65];
  __shared__ float sb[64];
  __shared__ float sw[64];                 // w2b pre-scaled by SELU_SCALE

  const int tid = threadIdx.x;
  const int i0  = blockIdx.x * 16;
  const int j0  = blockIdx.y * 16;

  for (int idx = tid; idx < 16 * 64; idx += 256) {
    const int r = idx >> 6, c = idx & 63;
    gi[r][c] = g[(size_t)(i0 + r) * 64 + c];
    gj[r][c] = g[(size_t)(j0 + r) * 64 + c];
  }
  if (tid < 64) {
    sb[tid] = b2a[tid];
    sw[tid] = w2b[tid] * SELU_SCALE;
  }
  __syncthreads();

  const int ti = tid >> 4;                 // i within tile
  const int tj = tid & 15;                 // j within tile (coalesced stores)

  float acc = 0.0f;
#pragma unroll
  for (int h = 0; h < 64; ++h) {
    const float pre = gi[ti][h] + gj[tj][h] + sb[h];
    const float s   = pre > 0.0f ? pre : SELU_ALPHA * (__expf(pre) - 1.0f);
    acc = fmaf(s, sw[h], acc);
  }

  float o = acc + b2b[0];
  o = o * (1.0f / 6.0f) + 0.5f;            // hardsigmoid
  o = fminf(fmaxf(o, 0.0f), 1.0f);

  const int i = i0 + ti, j = j0 + tj;
  out[(size_t)i * n + j] = o;
  out[(size_t)j * n + i] = o;              // mirror (same value when i==j)
}

extern "C" void kernel_launch(void* const* d_in, const int* in_sizes, int n_in,
                              void* d_out, int out_size, void* d_ws, size_t ws_size,
                              hipStream_t stream) {
  const float* x     = (const float*)d_in[0];   // (1536, 512)
  const float* W_enc = (const float*)d_in[1];   // (128, 512)
  const float* b_enc = (const float*)d_in[2];   // (128,)
  const float* W1a   = (const float*)d_in[3];   // (64, 128)
  const float* b1a   = (const float*)d_in[4];   // (64,)
  const float* W1b   = (const float*)d_in[5];   // (128, 64)
  const float* b1b   = (const float*)d_in[6];   // (128,)
  const float* W2a   = (const float*)d_in[7];   // (64, 128)
  const float* b2a   = (const float*)d_in[8];   // (64,)
  const float* W2b   = (const float*)d_in[9];   // (1, 64)
  const float* b2b   = (const float*)d_in[10];  // (1,)
  float* out = (float*)d_out;                   // (1536, 1536)

  const int N = 1536, RAW = 512, D = 128, H = 64;

  float* z  = (float*)d_ws;        // N*D
  float* t  = z  + (size_t)N * D;  // N*H
  float* hh = t  + (size_t)N * H;  // N*D
  float* g  = hh + (size_t)N * D;  // N*H   (total ~2.3 MB of workspace)

  const dim3 wave(32);
  wmma_gemm_nt<0><<<dim3(N / 16, D / 16), wave, 0, stream>>>(x,  W_enc, b_enc, z,  N, RAW, D);
  wmma_gemm_nt<1><<<dim3(N / 16, H / 16), wave, 0, stream>>>(z,  W1a,   b1a,   t,  N, D,   H);
  wmma_gemm_nt<1><<<dim3(N / 16, D / 16), wave, 0, stream>>>(t,  W1b,   b1b,   hh, N, H,   D);
  wmma_gemm_nt<0><<<dim3(N / 16, H / 16), wave, 0, stream>>>(hh, W2a,   nullptr, g, N, D,  H);

  pairwise_kernel<<<dim3(N / 16, N / 16), dim3(256), 0, stream>>>(g, b2a, W2b, b2b, out, N);
}